// KalmanFilter_74431783240430
// MI455X (gfx1250) — compile-verified
//
#include <hip/hip_runtime.h>

// ---------------------------------------------------------------------------
// Kalman filter, D=64, T=64, B=16384, fp32 end-to-end.
//
//   x_{t+1} = x_t @ A_t + y_t @ B_t
//   A_t = ((I - K_t H) F)^T,  B_t = K_t^T        (data-independent)
//
// Kernel 1 (1 WG, 256 thr): sequential P/S/K recursion. All 64x64 matmuls via
//   V_WMMA_F32_16X16X4_F32 (8 waves x 2 tiles); S^{-1} via register-blocked
//   Gauss-Jordan (each thread owns a 4x4 block of [S|SI] in VGPRs, pivot row
//   broadcast through small LDS buffers). Emits A_t/B_t packed in the WMMA
//   B-operand lane layout -> d_ws.
// Kernel 2: batched scan; 16 batch rows per wave, x tile resident in LDS,
//   V_WMMA_F32_16X16X4_F32 for the GEMMs, NT stores for the output.
// Needs ws_size >= 2 MiB (2 * 64 * 4096 floats).
// ---------------------------------------------------------------------------

typedef float v2f __attribute__((ext_vector_type(2)));
typedef float v4f __attribute__((ext_vector_type(4)));
typedef float v8f __attribute__((ext_vector_type(8)));

#define KF_D 64
#define KF_T 64
#define KF_B 16384

// ---------------------------------------------------------------------------
// Wave-tiled 64x64 WMMA matmul for the precompute WG (256 threads = 8 waves).
//   MODE 0: C = A * op(B)
//   MODE 1: C = A * op(B) + base
//   MODE 2: C = base - A * op(B)
//   BT:     op(B) = B^T (contiguous b64 operand loads), else B (paired b32)
// Uniform control flow (EXEC all ones). Caller must __syncthreads() after.
// ---------------------------------------------------------------------------
template <bool BT, int MODE>
__device__ __forceinline__ void wmma64(const float* __restrict__ A,
                                       const float* __restrict__ B,
                                       float* __restrict__ C,
                                       const float* __restrict__ base) {
  const int lane = threadIdx.x & 31;
  const int w    = threadIdx.x >> 5;  // 0..7
  const int m    = lane & 15;
  const int h    = lane >> 4;
  const v8f z = {0.f, 0.f, 0.f, 0.f, 0.f, 0.f, 0.f, 0.f};
#pragma unroll
  for (int i = 0; i < 2; ++i) {
    const int tile = w * 2 + i;  // 16 output tiles of 16x16
    const int mt = tile >> 2, nt = tile & 3;
    v8f acc = z;
#pragma unroll
    for (int k = 0; k < 16; ++k) {
      v2f a = *(const v2f*)(A + (16 * mt + m) * 64 + 4 * k + 2 * h);
      v2f b;
      if (BT) {
        b = *(const v2f*)(B + (16 * nt + m) * 64 + 4 * k + 2 * h);
      } else {
        b[0] = B[(4 * k + 2 * h) * 64 + 16 * nt + m];
        b[1] = B[(4 * k + 2 * h + 1) * 64 + 16 * nt + m];
      }
      acc = __builtin_amdgcn_wmma_f32_16x16x4_f32(false, a, false, b, (short)0,
                                                  acc, false, false);
    }
#pragma unroll
    for (int r = 0; r < 8; ++r) {
      const int row = 16 * mt + r + 8 * h;
      const int col = 16 * nt + m;
      float v = acc[r];
      if (MODE == 1) v = v + base[row * 64 + col];
      if (MODE == 2) v = base[row * 64 + col] - v;
      C[row * 64 + col] = v;
    }
  }
}

// ---------------------------- kernel 1: P/K recursion -----------------------

__global__ __launch_bounds__(256) void kf_precompute(
    const float* __restrict__ Fg, const float* __restrict__ Qg,
    const float* __restrict__ Hg, const float* __restrict__ Rg,
    const float* __restrict__ P0g, float* __restrict__ wsA,
    float* __restrict__ wsB) {
  extern __shared__ float smem[];
  float* P      = smem;            // 4096
  float* P1     = smem + 4096;     // 4096
  float* Sb     = smem + 8192;     // 4096 (S, later M = (I-KH)F)
  float* SIb    = smem + 12288;    // 4096
  float* Kb     = smem + 16384;    // 4096
  float* Wb     = smem + 20480;    // 4096
  float* prowS  = smem + 24576;    // 64: pivot row of S
  float* prowSI = smem + 24640;    // 64: pivot row of SI
  float* fbuf   = smem + 24704;    // 64: multiplier column

  const int tid = threadIdx.x;
  const int r0 = (tid >> 4) * 4;   // 4 owned rows
  const int c0 = (tid & 15) * 4;   // 4 owned cols

  for (int e = tid; e < KF_D * KF_D; e += blockDim.x) P[e] = P0g[e];
  __syncthreads();

  for (int t = 0; t < KF_T; ++t) {
    // P1 = F P F^T + Q
    wmma64<false, 0>(Fg, P, Wb, nullptr);  __syncthreads();
    wmma64<true, 1>(Wb, Fg, P1, Qg);       __syncthreads();
    // S = H P1 H^T + R
    wmma64<false, 0>(Hg, P1, Wb, nullptr); __syncthreads();
    wmma64<true, 1>(Wb, Hg, Sb, Rg);       __syncthreads();
    // W = P1 H^T (needed after the inversion)
    wmma64<true, 0>(P1, Hg, Wb, nullptr);  __syncthreads();

    // ---- SI = inv(S): register-blocked Gauss-Jordan (S is SPD) ----
    {
      float s[4][4], si[4][4];
#pragma unroll
      for (int i = 0; i < 4; ++i) {
        v4f tv = *(const v4f*)(Sb + (r0 + i) * 64 + c0);
#pragma unroll
        for (int j = 0; j < 4; ++j) {
          s[i][j]  = tv[j];
          si[i][j] = (r0 + i == c0 + j) ? 1.0f : 0.0f;
        }
      }
      auto publish = [&](int pn) {
#pragma unroll
        for (int ii = 0; ii < 4; ++ii) {
          if (r0 + ii == pn) {
            v4f ts  = {s[ii][0],  s[ii][1],  s[ii][2],  s[ii][3]};
            v4f tsi = {si[ii][0], si[ii][1], si[ii][2], si[ii][3]};
            *(v4f*)(prowS + c0)  = ts;
            *(v4f*)(prowSI + c0) = tsi;
          }
        }
#pragma unroll
        for (int jj = 0; jj < 4; ++jj) {
          if (c0 + jj == pn) {
#pragma unroll
            for (int ii = 0; ii < 4; ++ii) fbuf[r0 + ii] = s[ii][jj];
          }
        }
      };
      publish(0);
      __syncthreads();
      for (int p = 0; p < 64; ++p) {
        const float pv = prowS[p];
        const v4f pS  = *(const v4f*)(prowS + c0);
        const v4f pSI = *(const v4f*)(prowSI + c0);
        const v4f fv  = *(const v4f*)(fbuf + r0);
        __syncthreads();  // all reads done; buffers may be overwritten
        const float inv = 1.0f / pv;
#pragma unroll
        for (int i = 0; i < 4; ++i) {
          if (r0 + i == p) {
#pragma unroll
            for (int j = 0; j < 4; ++j) { s[i][j] *= inv; si[i][j] *= inv; }
          } else {
            const float g = fv[i] * inv;
#pragma unroll
            for (int j = 0; j < 4; ++j) {
              s[i][j]  = fmaf(-g, pS[j],  s[i][j]);
              si[i][j] = fmaf(-g, pSI[j], si[i][j]);
            }
          }
        }
        if (p < 63) publish(p + 1);
        __syncthreads();
      }
#pragma unroll
      for (int i = 0; i < 4; ++i) {
        v4f tsi = {si[i][0], si[i][1], si[i][2], si[i][3]};
        *(v4f*)(SIb + (r0 + i) * 64 + c0) = tsi;
      }
    }
    __syncthreads();

    // K = (P1 H^T) SI
    wmma64<false, 0>(Wb, SIb, Kb, nullptr); __syncthreads();
    // W = K H
    wmma64<false, 0>(Kb, Hg, Wb, nullptr);  __syncthreads();
    // P = P1 - (KH) P1
    wmma64<false, 2>(Wb, P1, P, P1);        __syncthreads();
    // M = F - (KH) F   (A_t = M^T), stored in Sb
    wmma64<false, 2>(Wb, Fg, Sb, Fg);       __syncthreads();

    // Pack A_t/B_t in WMMA B-operand layout:
    // chunk (n,k): lane l = 16h + c, vgpr s -> A_t[4k+2h+s][16n+c]
    //            = M[(16n+c)*64 + (4k+2h+s)]   (transpose applied here)
    for (int idx = tid; idx < KF_D * KF_D; idx += blockDim.x) {
      int s_ = idx & 1;
      int l  = (idx >> 1) & 31;
      int ck = (idx >> 6) & 15;
      int n  = idx >> 10;
      int hh = l >> 4, cc = l & 15;
      int row = 16 * n + cc;
      int col = 4 * ck + 2 * hh + s_;
      wsA[t * 4096 + idx] = Sb[row * KF_D + col];
      wsB[t * 4096 + idx] = Kb[row * KF_D + col];
    }
    __syncthreads();
  }
}

// ---------------------------- kernel 2: batched scan ------------------------

__global__ __launch_bounds__(256) void kf_scan(
    const float* __restrict__ y, const float* __restrict__ se,
    const float* __restrict__ wsA, const float* __restrict__ wsB,
    float* __restrict__ out) {
  __shared__ float xbuf[8][16 * 64];  // per-wave private x tile, 32 KB

  const int lane = threadIdx.x & 31;
  const int w    = threadIdx.x >> 5;          // wave in block (wave32)
  const int m    = lane & 15;                 // A-operand row / D column
  const int h    = lane >> 4;                 // lane-half select
  const int b0   = (blockIdx.x * 8 + w) * 16; // 16 batch rows per wave
  float* xw = xbuf[w];

  // x0 = state_estimate^T broadcast to all 16 rows
  for (int i = lane; i < 16 * 64; i += 32) xw[i] = se[i & 63];

  for (int t = 0; t < KF_T; ++t) {
    v8f z = {0.f, 0.f, 0.f, 0.f, 0.f, 0.f, 0.f, 0.f};
    v8f acc[4] = {z, z, z, z};

    const float* wa = wsA + t * 4096;
    const float* wb = wsB + t * 4096;
    const float* yr = y + (((size_t)(b0 + m)) * 64 + t) * 64 + 2 * h;

#pragma unroll
    for (int k = 0; k < 16; ++k) {
      // A-operand chunks (16x4): lane m holds row m, vgprs hold K=4k+2h(+1)
      v2f xa = *(const v2f*)(xw + m * 64 + 4 * k + 2 * h);   // ds_load_b64
      v2f ya = *(const v2f*)(yr + 4 * k);                    // global b64
#pragma unroll
      for (int n = 0; n < 4; ++n) {
        v2f bA = *(const v2f*)(wa + n * 1024 + k * 64 + 2 * lane);
        v2f bB = *(const v2f*)(wb + n * 1024 + k * 64 + 2 * lane);
        acc[n] = __builtin_amdgcn_wmma_f32_16x16x4_f32(
            false, xa, false, bA, (short)0, acc[n], false, false);
        acc[n] = __builtin_amdgcn_wmma_f32_16x16x4_f32(
            false, ya, false, bB, (short)0, acc[n], false, false);
      }
    }

    // D layout: vgpr r, lane l<16 -> (row r,   col 16n+l)
    //                    lane>=16 -> (row r+8, col 16n+l-16)
#pragma unroll
    for (int n = 0; n < 4; ++n) {
#pragma unroll
      for (int r = 0; r < 8; ++r) {
        int row = r + 8 * h;
        int col = 16 * n + m;
        float val = acc[n][r];
        __builtin_nontemporal_store(
            val, &out[(((size_t)(b0 + row)) * 64 + t) * 64 + col]);
        xw[row * 64 + col] = val;  // feeds next step (same-wave DS ordering)
      }
    }
  }
}

// ------------------------------- launcher -----------------------------------

extern "C" void kernel_launch(void* const* d_in, const int* in_sizes, int n_in,
                              void* d_out, int out_size, void* d_ws,
                              size_t ws_size, hipStream_t stream) {
  (void)in_sizes; (void)n_in; (void)out_size; (void)ws_size;
  const float* y  = (const float*)d_in[0];  // [B, T, D]
  const float* F  = (const float*)d_in[1];
  const float* Q  = (const float*)d_in[2];
  const float* H  = (const float*)d_in[3];
  const float* R  = (const float*)d_in[4];
  const float* se = (const float*)d_in[5];  // [D, 1]
  const float* P0 = (const float*)d_in[6];
  float* out = (float*)d_out;

  float* wsA = (float*)d_ws;            // 64 * 4096 floats
  float* wsB = wsA + KF_T * KF_D * KF_D;

  kf_precompute<<<dim3(1), dim3(256), (6 * 4096 + 192) * sizeof(float),
                  stream>>>(F, Q, H, R, P0, wsA, wsB);

  kf_scan<<<dim3(KF_B / 128), dim3(256), 0, stream>>>(y, se, wsA, wsB, out);
}